// BaseModel_18081812316555
// MI455X (gfx1250) — compile-verified
//
#include <hip/hip_runtime.h>
#include <math.h>

// ---------------------------------------------------------------------------
// Problem constants (match reference)
// ---------------------------------------------------------------------------
#define BB    16
#define CN    22
#define TT    1000
#define EXP_  300
#define FCH   150
#define SPD   100
#define NCLS  4
#define NFLAT 5050      // SPD*(SPD+1)/2
#define HALO  37        // max temporal kernel (75) / 2

// Padded dims (so all WMMA GEMMs run with zero bounds checks)
#define T_P   1008      // T padded to 16
#define CH_P  480       // 3 blocks x 160 padded channels (rows 150..159 zero)
#define SPD_P 112       // SPD padded to 16
#define K_X   24        // electrode dim padded to mult of 4

typedef __attribute__((ext_vector_type(2))) float v2f;
typedef __attribute__((ext_vector_type(8))) float v8f;

// CDNA5 full-precision matrix op: D(16x16,f32) = A(16x4,f32) x B(4x16,f32) + C
__device__ __forceinline__ v8f wmma_f32_16x16x4(v2f a, v2f b, v8f c) {
  return __builtin_amdgcn_wmma_f32_16x16x4_f32(
      /*neg_a=*/false, a, /*neg_b=*/false, b,
      /*c_mod=*/(short)0, c, /*reuse_a=*/false, /*reuse_b=*/false);
}

// ---------------------------------------------------------------------------
// Checkless batched wave-tile GEMM on padded operands.
//   C[b] = alpha * A[b] * op(B[b]) (+ diagAdd on the diagonal)
// A: (Mpad x K) row-major lda; B: (K x Npad) ldb, or (Npad x K) if TRANSB.
// Grid covers padded M/N exactly; K is a multiple of 4. One wave per tile.
// sA/sB/sC = 0 -> operand shared across batch.
// ---------------------------------------------------------------------------
template <bool TRANSB>
__global__ void __launch_bounds__(32)
gemm_wmma(const float* __restrict__ A, long sA, int lda,
          const float* __restrict__ B, long sB, int ldb,
          float* __restrict__ C, long sC, int ldc,
          int K, float alpha, float diagAdd)
{
  const int lane = threadIdx.x;
  const int m0 = blockIdx.x * 16, n0 = blockIdx.y * 16, b = blockIdx.z;
  const int row   = m0 + (lane & 15);
  const int col   = n0 + (lane & 15);
  const int khalf = (lane >> 4) << 1;   // lanes 0-15 -> K offset 0, 16-31 -> 2

  const float* Arow = A + (long)b * sA + (long)row * lda + khalf;  // 8B aligned
  const float* Bb   = B + (long)b * sB;
  float*       Cb   = C + (long)b * sC;

  v8f acc = {};
#pragma unroll 4
  for (int kk = 0; kk < K; kk += 4) {
    v2f af = *reinterpret_cast<const v2f*>(Arow + kk);
    v2f bf;
    if (TRANSB) {
      bf = *reinterpret_cast<const v2f*>(Bb + (long)col * ldb + kk + khalf);
    } else {
      const float* bp = Bb + (long)(kk + khalf) * ldb + col;
      bf.x = bp[0];
      bf.y = bp[ldb];
      // speculative prefetch of B a few k-rows ahead (global_prefetch_b8;
      // dropped silently if the address translation fails)
      __builtin_prefetch(bp + 16 * (long)ldb, 0, 1);
    }
    acc = wmma_f32_16x16x4(af, bf, acc);
  }

  const int nl = lane & 15, mb = (lane >> 4) * 8;
#pragma unroll
  for (int i = 0; i < 8; ++i) {
    const int m = m0 + mb + i, n = n0 + nl;
    float vv = alpha * acc[i];
    if (m == n) vv += diagAdd;
    Cb[(long)m * ldc + n] = vv;
  }
}

// ---------------------------------------------------------------------------
// Pack kernels: zero-padded operand staging (padding is exactly zero)
// ---------------------------------------------------------------------------
__global__ void __launch_bounds__(256)
pack_x(const float* __restrict__ x, float* __restrict__ Xp) // (16,22,1000)->(16,24,1008)
{
  const int b = blockIdx.x;
  for (int e = threadIdx.x; e < K_X * T_P; e += 256) {
    const int c = e / T_P, t = e % T_P;
    Xp[(long)b * K_X * T_P + e] =
        (c < CN && t < TT) ? x[((long)b * CN + c) * TT + t] : 0.f;
  }
}

__global__ void __launch_bounds__(256)
pack_wsconv(const float* __restrict__ w, float* __restrict__ Wp) // (100,450)->(112,480)
{
  for (int e = blockIdx.x * 256 + threadIdx.x; e < SPD_P * CH_P;
       e += gridDim.x * 256) {
    const int p = e / CH_P, cc = e % CH_P;
    const int blk = cc / 160, o = cc % 160;
    Wp[e] = (p < SPD && o < FCH) ? w[p * (3 * FCH) + blk * FCH + o] : 0.f;
  }
}

__global__ void __launch_bounds__(256)
pack_aff(const float* __restrict__ w, float* __restrict__ Wa) // (100,100)->(112,112)
{
  for (int e = threadIdx.x; e < SPD_P * SPD_P; e += 256) {
    const int i = e / SPD_P, j = e % SPD_P;
    Wa[e] = (i < SPD && j < SPD) ? w[i * SPD + j] : 0.f;
  }
}

// ---------------------------------------------------------------------------
// Fold w1 (3-tap electrode conv) + w2 (expand) + max-norm(w3) (spatial
// collapse) + w4 (pointwise) into a single 150x22 matrix per block,
// stored zero-padded to 160x24 for WMMA tiling.
// ---------------------------------------------------------------------------
__global__ void __launch_bounds__(256)
prep_wfold(const float* __restrict__ w1, const float* __restrict__ w2,
           const float* __restrict__ w3, const float* __restrict__ w4,
           float* __restrict__ out /* [160][24] zero-padded */)
{
  __shared__ float s[EXP_];
  __shared__ float w3s[EXP_ * CN];
  __shared__ float weff[FCH * CN];
  const int tid = threadIdx.x;

  for (int e = tid; e < 160 * K_X; e += 256) out[e] = 0.f;

  for (int g = tid; g < EXP_; g += 256) {
    float nrm = 0.f;
    for (int c = 0; c < CN; ++c) {
      float v = w3[g * CN + c];
      w3s[g * CN + c] = v;
      nrm += v * v;
    }
    nrm = sqrtf(nrm);
    s[g] = w2[g] * fminf(1.f, 1.f / (nrm + 1e-12f));   // max-norm constraint
  }
  __syncthreads();

  for (int e = tid; e < FCH * CN; e += 256) {
    const int o = e / CN, c = e % CN;
    float acc = 0.f;
    for (int g = 0; g < EXP_; ++g)
      acc += w4[o * EXP_ + g] * s[g] * w3s[g * CN + c];
    weff[e] = acc;
  }
  __syncthreads();

  const float k0 = w1[0], k1 = w1[1], k2 = w1[2];
  for (int e = tid; e < FCH * CN; e += 256) {
    const int o = e / CN, c = e % CN;
    float acc = k1 * weff[o * CN + c];
    if (c + 1 < CN) acc += k0 * weff[o * CN + c + 1];
    if (c - 1 >= 0) acc += k2 * weff[o * CN + c - 1];
    out[o * K_X + c] = acc;
  }
}

// ---------------------------------------------------------------------------
// Front GEMM (checkless): H[b, blk*160+o, t] = sum_c Wfold[blk][o][c]*Xp[b][c][t]
// Wfold rows 150..159 are zero -> H padding rows are exact zeros.
// ---------------------------------------------------------------------------
__global__ void __launch_bounds__(32)
front_gemm(const float* __restrict__ Wfold /*[3][160][24]*/,
           const float* __restrict__ Xp    /*[16][24][1008]*/,
           float* __restrict__ H           /*[16][480][1008]*/)
{
  const int lane = threadIdx.x;
  const int m0 = blockIdx.x * 16, n0 = blockIdx.y * 16;
  const int blk = blockIdx.z >> 4, b = blockIdx.z & 15;
  const int row   = m0 + (lane & 15);
  const int col   = n0 + (lane & 15);
  const int khalf = (lane >> 4) << 1;

  const float* Arow = Wfold + blk * 160 * K_X + row * K_X + khalf;
  const float* X    = Xp + (long)b * K_X * T_P;

  v8f acc = {};
#pragma unroll
  for (int kk = 0; kk < K_X; kk += 4) {
    v2f af = *reinterpret_cast<const v2f*>(Arow + kk);
    v2f bf;
    const float* bp = X + (long)(kk + khalf) * T_P + col;
    bf.x = bp[0];
    bf.y = bp[T_P];
    acc = wmma_f32_16x16x4(af, bf, acc);
  }
  const int nl = lane & 15, mb = (lane >> 4) * 8;
#pragma unroll
  for (int i = 0; i < 8; ++i) {
    const int o = m0 + mb + i, t = n0 + nl;
    H[((long)b * CH_P + blk * 160 + o) * T_P + t] = acc[i];
  }
}

// ---------------------------------------------------------------------------
// Block reduction helper
// ---------------------------------------------------------------------------
__device__ __forceinline__ float block_sum(float v, float* red) {
  const int tid = threadIdx.x;
  red[tid] = v; __syncthreads();
  for (int s = 128; s > 0; s >>= 1) {
    if (tid < s) red[tid] += red[tid + s];
    __syncthreads();
  }
  float r = red[0];
  __syncthreads();
  return r;
}

// ---------------------------------------------------------------------------
// LN(T) -> depthwise temporal conv (same, zero pad) -> LN(T), in-place on a
// 1000-sample channel row. One 256-thread block per (b, channel) row.
// Only touches t < 1000; padded columns stay zero.
// ---------------------------------------------------------------------------
__global__ void __launch_bounds__(256)
row_post(float* __restrict__ H, int blk, int K,
         const float* __restrict__ g1, const float* __restrict__ b1,
         const float* __restrict__ w5,
         const float* __restrict__ g2, const float* __restrict__ b2)
{
  __shared__ float u[TT + 2 * HALO];
  __shared__ float vbuf[TT];
  __shared__ float wk[80];
  __shared__ float red[256];
  const int tid = threadIdx.x;
  const int b = blockIdx.x / FCH;
  const int o = blockIdx.x % FCH;
  float* row = H + ((long)b * CH_P + blk * 160 + o) * T_P;

  for (int t = tid; t < TT + 2 * HALO; t += 256) u[t] = 0.f;
  for (int j = tid; j < K; j += 256) wk[j] = w5[o * K + j];
  __syncthreads();

  float ps = 0.f, pq = 0.f;
  for (int t = tid; t < TT; t += 256) {
    float vv = row[t];
    u[HALO + t] = vv;
    ps += vv; pq += vv * vv;
  }
  const float mean = block_sum(ps, red) * (1.f / TT);
  const float var  = block_sum(pq, red) * (1.f / TT) - mean * mean;
  const float rstd = rsqrtf(var + 1e-5f);
  for (int t = tid; t < TT; t += 256)
    u[HALO + t] = (u[HALO + t] - mean) * rstd * g1[t] + b1[t];
  __syncthreads();

  const int half = K >> 1;
  float ps2 = 0.f, pq2 = 0.f;
  for (int t = tid; t < TT; t += 256) {
    float acc = 0.f;
    for (int j = 0; j < K; ++j) acc += wk[j] * u[HALO + t + j - half];
    vbuf[t] = acc;
    ps2 += acc; pq2 += acc * acc;
  }
  const float mean2 = block_sum(ps2, red) * (1.f / TT);
  const float var2  = block_sum(pq2, red) * (1.f / TT) - mean2 * mean2;
  const float rstd2 = rsqrtf(var2 + 1e-5f);
  for (int t = tid; t < TT; t += 256)
    row[t] = (vbuf[t] - mean2) * rstd2 * g2[t] + b2[t];
}

// Center each S row over t<1000 (padded columns stay exactly zero)
__global__ void __launch_bounds__(256)
center_rows(float* __restrict__ S)
{
  __shared__ float red[256];
  const int tid = threadIdx.x;
  float* row = S + (long)blockIdx.x * T_P;
  float ps = 0.f;
  for (int t = tid; t < TT; t += 256) ps += row[t];
  const float mean = block_sum(ps, red) * (1.f / TT);
  for (int t = tid; t < TT; t += 256) row[t] -= mean;
}

// ---------------------------------------------------------------------------
// Batched symmetric eigendecomposition: cyclic Jacobi with round-robin
// parallel ordering (50 disjoint pairs / round, 99 rounds / sweep).
// One workgroup per 100x100 SPD matrix (read from ld=112 padded P).
// A in LDS; V (112x112, padded region zero) in global ws.
// ---------------------------------------------------------------------------
#define LDA 101
__global__ void __launch_bounds__(256)
jacobi_eigh(const float* __restrict__ P, float* __restrict__ V,
            float* __restrict__ lw)
{
  __shared__ float Am[SPD * LDA];
  __shared__ int   pp[SPD / 2];
  __shared__ int   qq[SPD / 2];
  __shared__ float cs[SPD / 2];
  __shared__ float sn[SPD / 2];
  const int tid = threadIdx.x;
  const int b = blockIdx.x;
  const float* Pb = P + (long)b * SPD_P * SPD_P;
  float*       Vb = V + (long)b * SPD_P * SPD_P;

  for (int e = tid; e < SPD * SPD; e += 256)
    Am[(e / SPD) * LDA + (e % SPD)] = Pb[(e / SPD) * SPD_P + (e % SPD)];
  for (int e = tid; e < SPD_P * SPD_P; e += 256) {
    const int i = e / SPD_P, j = e % SPD_P;
    Vb[e] = (i == j && i < SPD) ? 1.f : 0.f;
  }
  __syncthreads();

  const int nm1 = SPD - 1;
  for (int sweep = 0; sweep < 10; ++sweep) {
    for (int r = 0; r < nm1; ++r) {
      if (tid < SPD / 2) {
        int p, q;
        if (tid == 0) { p = nm1; q = r % nm1; }
        else { p = (r + tid) % nm1; q = (r - tid + nm1) % nm1; }
        if (p > q) { int t0 = p; p = q; q = t0; }
        const float app = Am[p * LDA + p];
        const float aqq = Am[q * LDA + q];
        const float apq = Am[p * LDA + q];
        float c = 1.f, s = 0.f;
        if (fabsf(apq) > 1e-12f) {
          const float tau = (aqq - app) / (2.f * apq);
          const float t = (tau >= 0.f) ? 1.f / (tau + sqrtf(1.f + tau * tau))
                                       : 1.f / (tau - sqrtf(1.f + tau * tau));
          c = rsqrtf(1.f + t * t);
          s = t * c;
        }
        pp[tid] = p; qq[tid] = q; cs[tid] = c; sn[tid] = s;
      }
      __syncthreads();
      // Row phase: A <- J^T A  (pairs own disjoint row sets)
      for (int w = tid; w < (SPD / 2) * SPD; w += 256) {
        const int pr = w / SPD, k = w % SPD;
        const int p = pp[pr], q = qq[pr];
        const float c = cs[pr], s = sn[pr];
        const float apk = Am[p * LDA + k], aqk = Am[q * LDA + k];
        Am[p * LDA + k] = c * apk - s * aqk;
        Am[q * LDA + k] = s * apk + c * aqk;
      }
      __syncthreads();
      // Col phase: A <- A J ; V <- V J  (pairs own disjoint column sets)
      for (int w = tid; w < (SPD / 2) * SPD; w += 256) {
        const int pr = w / SPD, k = w % SPD;
        const int p = pp[pr], q = qq[pr];
        const float c = cs[pr], s = sn[pr];
        const float akp = Am[k * LDA + p], akq = Am[k * LDA + q];
        Am[k * LDA + p] = c * akp - s * akq;
        Am[k * LDA + q] = s * akp + c * akq;
        const float vkp = Vb[k * SPD_P + p], vkq = Vb[k * SPD_P + q];
        Vb[k * SPD_P + p] = c * vkp - s * vkq;
        Vb[k * SPD_P + q] = s * vkp + c * vkq;
      }
      __syncthreads();
    }
  }
  for (int j = tid; j < SPD; j += 256)
    lw[b * SPD + j] = logf(fmaxf(Am[j * LDA + j], 1e-6f));
}

// Vw[b][i][k] = V[b][i][k] * lw[b][k] for k<SPD, 0 in padding (L = Vw @ V^T)
__global__ void __launch_bounds__(256)
scale_cols(const float* __restrict__ V, const float* __restrict__ lw,
           float* __restrict__ Vw)
{
  const int b = blockIdx.x;
  for (int e = threadIdx.x; e < SPD_P * SPD_P; e += 256) {
    const int k = e % SPD_P;
    Vw[(long)b * SPD_P * SPD_P + e] =
        (k < SPD) ? V[(long)b * SPD_P * SPD_P + e] * lw[b * SPD + k] : 0.f;
  }
}

// Upper-triangular half-vectorization with sqrt(2) off-diagonal scaling
__global__ void __launch_bounds__(256)
vec_flat(const float* __restrict__ L, float* __restrict__ out_flat)
{
  const int b = blockIdx.x;
  const float* Lb = L + (long)b * SPD_P * SPD_P;
  const float s2 = 1.41421356237309515f;
  for (int w = threadIdx.x; w < NFLAT; w += 256) {
    int i = (int)((2 * SPD + 1 -
                   sqrtf((float)(2 * SPD + 1) * (2 * SPD + 1) - 8.f * (float)w)) * 0.5f);
    if (i < 0) i = 0;
    if (i > SPD - 1) i = SPD - 1;
    while (i > 0 && (i * SPD - i * (i - 1) / 2) > w) --i;
    while (((i + 1) * SPD - (i + 1) * i / 2) <= w) ++i;
    const int base = i * SPD - i * (i - 1) / 2;
    const int j = i + (w - base);
    out_flat[(long)b * NFLAT + w] = Lb[i * SPD_P + j] * ((i == j) ? 1.f : s2);
  }
}

// logits[b][c] = flat[b] . w_fc[c] + b_fc[c]
__global__ void __launch_bounds__(256)
fc_logits(const float* __restrict__ flat, const float* __restrict__ w_fc,
          const float* __restrict__ b_fc, float* __restrict__ logits)
{
  __shared__ float red[256];
  const int b = blockIdx.x >> 2, c = blockIdx.x & 3;
  float ps = 0.f;
  for (int idx = threadIdx.x; idx < NFLAT; idx += 256)
    ps += flat[(long)b * NFLAT + idx] * w_fc[c * NFLAT + idx];
  const float tot = block_sum(ps, red);
  if (threadIdx.x == 0) logits[b * NCLS + c] = tot + b_fc[c];
}

// ---------------------------------------------------------------------------
// Host launcher
// ---------------------------------------------------------------------------
extern "C" void kernel_launch(void* const* d_in, const int* in_sizes, int n_in,
                              void* d_out, int out_size, void* d_ws, size_t ws_size,
                              hipStream_t stream) {
  const float* x = (const float*)d_in[0];
  const float* bw1[3] = {(const float*)d_in[1],  (const float*)d_in[10], (const float*)d_in[19]};
  const float* bw2[3] = {(const float*)d_in[2],  (const float*)d_in[11], (const float*)d_in[20]};
  const float* bw3[3] = {(const float*)d_in[3],  (const float*)d_in[12], (const float*)d_in[21]};
  const float* bw4[3] = {(const float*)d_in[4],  (const float*)d_in[13], (const float*)d_in[22]};
  const float* bg1[3] = {(const float*)d_in[5],  (const float*)d_in[14], (const float*)d_in[23]};
  const float* bb1[3] = {(const float*)d_in[6],  (const float*)d_in[15], (const float*)d_in[24]};
  const float* bw5[3] = {(const float*)d_in[7],  (const float*)d_in[16], (const float*)d_in[25]};
  const float* bg2[3] = {(const float*)d_in[8],  (const float*)d_in[17], (const float*)d_in[26]};
  const float* bb2[3] = {(const float*)d_in[9],  (const float*)d_in[18], (const float*)d_in[27]};
  const float* w_sconv = (const float*)d_in[28];
  const float* w_aff   = (const float*)d_in[29];
  const float* w_fc    = (const float*)d_in[30];
  const float* b_fc    = (const float*)d_in[31];
  const int Ks[3] = {15, 75, 55};

  const long SQ = (long)SPD_P * SPD_P;   // 12544
  float* ws  = (float*)d_ws;
  float* WF  = ws;                        // 3*160*24        = 11520
  float* Xp  = WF  + 3 * 160 * K_X;       // 16*24*1008      = 387072
  float* H   = Xp  + 16L * K_X * T_P;     // 16*480*1008     = 7741440
  float* S   = H   + 16L * CH_P * T_P;    // 16*112*1008     = 1806336
  float* Wp  = S   + 16L * SPD_P * T_P;   // 112*480         = 53760
  float* Wa  = Wp  + (long)SPD_P * CH_P;  // 112*112         = 12544
  float* COV = Wa  + SQ;                  // 16*12544
  float* TMP = COV + 16 * SQ;
  float* Pm  = TMP + 16 * SQ;
  float* V   = Pm  + 16 * SQ;
  float* LW  = V   + 16 * SQ;             // 16*100
  float* VW  = LW  + 16L * SPD;
  float* Lm  = VW  + 16 * SQ;             // end ~11.2M floats (~45 MB)

  float* logits = (float*)d_out;               // [16][4]
  float* flat   = (float*)d_out + BB * NCLS;   // [16][5050]

  // 0) stage padded operands
  pack_x<<<dim3(BB), 256, 0, stream>>>(x, Xp);
  pack_wsconv<<<dim3(210), 256, 0, stream>>>(w_sconv, Wp);
  pack_aff<<<dim3(1), 256, 0, stream>>>(w_aff, Wa);

  // 1) fold each block's front (w1,w2,norm(w3),w4) into a 150x22 matrix
  for (int blk = 0; blk < 3; ++blk)
    prep_wfold<<<1, 256, 0, stream>>>(bw1[blk], bw2[blk], bw3[blk], bw4[blk],
                                      WF + blk * 160 * K_X);

  // 2) front GEMM (WMMA f32, checkless): H = Wfold @ Xp
  front_gemm<<<dim3(10, 63, 48), 32, 0, stream>>>(WF, Xp, H);

  // 3) LN -> depthwise temporal conv -> LN, in place on H
  for (int blk = 0; blk < 3; ++blk)
    row_post<<<dim3(BB * FCH), 256, 0, stream>>>(H, blk, Ks[blk],
        bg1[blk], bb1[blk], bw5[blk], bg2[blk], bb2[blk]);

  // 4) spatial conv: S[b] = Wp (112x480) @ H[b] (480x1008)
  gemm_wmma<false><<<dim3(7, 63, 16), 32, 0, stream>>>(
      Wp, 0, CH_P, H, (long)CH_P * T_P, T_P,
      S, (long)SPD_P * T_P, T_P, CH_P, 1.f, 0.f);

  // 5) center over time
  center_rows<<<dim3(BB * SPD_P), 256, 0, stream>>>(S);

  // 6) covariance: COV[b] = Sc Sc^T / (T-1) + 1e-4 I  (padding cols are zero)
  gemm_wmma<true><<<dim3(7, 7, 16), 32, 0, stream>>>(
      S, (long)SPD_P * T_P, T_P, S, (long)SPD_P * T_P, T_P,
      COV, SQ, SPD_P, T_P, 1.f / (TT - 1), 1e-4f);

  // 7) congruence: P = Wa @ COV @ Wa^T (two WMMA GEMMs)
  gemm_wmma<false><<<dim3(7, 7, 16), 32, 0, stream>>>(
      Wa, 0, SPD_P, COV, SQ, SPD_P,
      TMP, SQ, SPD_P, SPD_P, 1.f, 0.f);
  gemm_wmma<true><<<dim3(7, 7, 16), 32, 0, stream>>>(
      TMP, SQ, SPD_P, Wa, 0, SPD_P,
      Pm, SQ, SPD_P, SPD_P, 1.f, 0.f);

  // 8) batched Jacobi eigh -> V, log-eigenvalues
  jacobi_eigh<<<dim3(BB), 256, 0, stream>>>(Pm, V, LW);

  // 9) L = V diag(lw) V^T  (scale columns, then WMMA GEMM with B^T)
  scale_cols<<<dim3(BB), 256, 0, stream>>>(V, LW, VW);
  gemm_wmma<true><<<dim3(7, 7, 16), 32, 0, stream>>>(
      VW, SQ, SPD_P, V, SQ, SPD_P,
      Lm, SQ, SPD_P, SPD_P, 1.f, 0.f);

  // 10) half-vectorize into d_out, then logits
  vec_flat<<<dim3(BB), 256, 0, stream>>>(Lm, flat);
  fc_logits<<<dim3(BB * NCLS), 256, 0, stream>>>(flat, w_fc, b_fc, logits);
}